// Window_40939628266118
// MI455X (gfx1250) — compile-verified
//
#include <hip/hip_runtime.h>
#include <hip/hip_bf16.h>

typedef float v2f __attribute__((ext_vector_type(2)));
typedef float v8f __attribute__((ext_vector_type(8)));

#define B_    2048
#define CELL_ 512
#define K_    30
#define L_    1024
#define V_    80

// ---------------------------------------------------------------------------
// Kernel 1: params = exp(x @ W^T + bias)  via V_WMMA_F32_16X16X4_F32
//           alpha = params[:, 0:30], beta = params[:, 30:60],
//           kappa = kappa_old + params[:, 60:90]
// One wave (32 threads) computes a 16(M) x 16(N) tile, K looped in steps of 4.
// A (16x4 f32): lanes 0-15 -> M, v0 = K(2*half), v1 = K(2*half+1)
// B (4x16 f32): lanes 0-15 -> N, v0 = K(2*half), v1 = K(2*half+1)
// C/D (16x16):  VGPR r, lanes 0-15 -> M=r, lanes 16-31 -> M=8+r, N = lane&15
// B loads for padded columns (j >= 90) come from a CLAMPED row pointer and are
// zeroed with a lane select (v_cndmask) -- no EXEC divergence in the WMMA loop.
// ---------------------------------------------------------------------------
__global__ __launch_bounds__(32) void win_gemm_wmma(
    const float* __restrict__ x, const float* __restrict__ W,
    const float* __restrict__ bias, const float* __restrict__ kappa_old,
    float* __restrict__ alpha, float* __restrict__ beta,
    float* __restrict__ kappa_ws, float* __restrict__ kappa_out)
{
    const int lane = threadIdx.x;
    const int lm   = lane & 15;
    const int half = lane >> 4;
    const int m0   = blockIdx.x * 16;
    const int n0   = blockIdx.y * 16;
    const int j    = n0 + lm;                 // output column (0..95, valid < 90)
    const bool valid = (j < 90);
    const int jc   = valid ? j : 89;          // clamped row for safe loads

    const float* xrow = x + (size_t)(m0 + lm) * CELL_ + 2 * half;
    const float* wrow = W + (size_t)jc * CELL_ + 2 * half;

    v8f c = {};
    #pragma unroll 4
    for (int k0 = 0; k0 < CELL_; k0 += 4) {
        v2f a = *(const v2f*)(xrow + k0);
        v2f t = *(const v2f*)(wrow + k0);     // unconditional, clamped address
        v2f bb;
        bb[0] = valid ? t[0] : 0.0f;          // lane select, keeps EXEC full
        bb[1] = valid ? t[1] : 0.0f;
        // (neg_a, A, neg_b, B, c_mod, C, reuse_a, reuse_b)
        c = __builtin_amdgcn_wmma_f32_16x16x4_f32(
                false, a, false, bb, (short)0, c, false, false);
    }

    if (valid) {
        const float bj = bias[j];
        #pragma unroll
        for (int r = 0; r < 8; ++r) {
            const int row = m0 + r + 8 * half;
            const float val = __expf(c[r] + bj);
            if (j < 30) {
                alpha[row * K_ + j] = val;
            } else if (j < 60) {
                beta[row * K_ + (j - 30)] = val;
            } else {
                const int kk = j - 60;
                const float kp = kappa_old[row * K_ + kk] + val;
                kappa_ws[row * K_ + kk]  = kp;
                kappa_out[row * K_ + kk] = kp;
            }
        }
    }
}

// ---------------------------------------------------------------------------
// Kernel 2: phi[b,l] = (L / text_len[b]) * sum_k alpha[b,k] *
//                      exp(-beta[b,k] * (kappa[b,k] - l)^2),  l in [0, L]
// One block per b; alpha/beta/kappa staged in LDS; v_exp_f32 inner loop.
// ---------------------------------------------------------------------------
__global__ __launch_bounds__(256) void win_phi(
    const float* __restrict__ alpha, const float* __restrict__ beta,
    const float* __restrict__ kappa, const float* __restrict__ text_lens,
    float* __restrict__ phi)
{
    __shared__ float sa[K_], sb[K_], sk[K_];
    const int b   = blockIdx.x;
    const int tid = threadIdx.x;

    if (tid < K_)            sa[tid]          = alpha[b * K_ + tid];
    else if (tid < 2 * K_)   sb[tid - K_]     = beta [b * K_ + (tid - K_)];
    else if (tid < 3 * K_)   sk[tid - 2 * K_] = kappa[b * K_ + (tid - 2 * K_)];
    __syncthreads();

    const float scale = (float)L_ / text_lens[b];
    for (int l = tid; l <= L_; l += 256) {
        const float fl = (float)l;
        float s = 0.0f;
        #pragma unroll
        for (int k = 0; k < K_; ++k) {
            const float d = sk[k] - fl;
            s += sa[k] * __expf(-sb[k] * d * d);
        }
        phi[b * (L_ + 1) + l] = s * scale;
    }
}

// ---------------------------------------------------------------------------
// Kernel 3 (bandwidth kernel): w[b,v] = sum_l phi[b,l] * onehots[b,l,v]
// One 320-thread block per b. Per step: 16 rows x 20 float4 = 5120 contiguous
// bytes, fully coalesced. 671 MB streamed exactly once -> ~29 us HBM floor.
// ---------------------------------------------------------------------------
__global__ __launch_bounds__(320) void win_w(
    const float* __restrict__ phi, const float* __restrict__ onehots,
    float* __restrict__ w)
{
    __shared__ float4 s[320];
    const int b   = blockIdx.x;
    const int tid = threadIdx.x;
    const int r   = tid / 20;   // 0..15: row within 16-row step
    const int j   = tid % 20;   // float4 column (covers v = 4j..4j+3)

    const float4* oh = (const float4*)(onehots + (size_t)b * (L_ * V_));
    const float*  ph = phi + (size_t)b * (L_ + 1);

    float4 acc = make_float4(0.0f, 0.0f, 0.0f, 0.0f);
    for (int l0 = 0; l0 < L_; l0 += 16) {
        const int l = l0 + r;
        if (l0 + 16 < L_)
            __builtin_prefetch(&oh[(size_t)(l + 16) * 20 + j], 0, 1);
        const float  p = ph[l];
        const float4 o = oh[(size_t)l * 20 + j];
        acc.x += p * o.x; acc.y += p * o.y;
        acc.z += p * o.z; acc.w += p * o.w;
    }

    s[tid] = acc;
    __syncthreads();

    if (tid < V_) {
        const int jj = tid >> 2, cc = tid & 3;
        float sum = 0.0f;
        #pragma unroll
        for (int rr = 0; rr < 16; ++rr) {
            const float* p4 = (const float*)&s[rr * 20 + jj];
            sum += p4[cc];
        }
        w[(size_t)b * V_ + tid] = sum;
    }
}

// ---------------------------------------------------------------------------
extern "C" void kernel_launch(void* const* d_in, const int* in_sizes, int n_in,
                              void* d_out, int out_size, void* d_ws, size_t ws_size,
                              hipStream_t stream)
{
    (void)in_sizes; (void)n_in; (void)out_size; (void)ws_size;

    const float* x         = (const float*)d_in[0];  // [B, CELL]
    const float* kappa_old = (const float*)d_in[1];  // [B, K]
    const float* onehots   = (const float*)d_in[2];  // [B, L, V]
    const float* text_lens = (const float*)d_in[3];  // [B, 1]
    const float* W         = (const float*)d_in[4];  // [3K, CELL]
    const float* bias      = (const float*)d_in[5];  // [3K]

    float* out       = (float*)d_out;
    float* w_out     = out;                          // [B, V]
    float* kappa_out = out + (size_t)B_ * V_;        // [B, K]
    float* phi_out   = kappa_out + (size_t)B_ * K_;  // [B, L+1]

    float* ws       = (float*)d_ws;                  // 3 * B * K floats
    float* alpha    = ws;
    float* beta     = ws + (size_t)B_ * K_;
    float* kappa_ws = ws + (size_t)2 * B_ * K_;

    win_gemm_wmma<<<dim3(B_ / 16, 6), 32, 0, stream>>>(
        x, W, bias, kappa_old, alpha, beta, kappa_ws, kappa_out);

    win_phi<<<B_, 256, 0, stream>>>(alpha, beta, kappa_ws, text_lens, phi_out);

    win_w<<<B_, 320, 0, stream>>>(phi_out, onehots, w_out);
}